// GCN_21320217658154
// MI455X (gfx1250) — compile-verified
//
#include <hip/hip_runtime.h>
#include <cstdint>

typedef float v2f __attribute__((ext_vector_type(2)));
typedef float v8f __attribute__((ext_vector_type(8)));

#define NEG_SLOPE 0.2f

__device__ __forceinline__ float warp_sum(float v) {
#pragma unroll
    for (int m = 16; m >= 1; m >>= 1) v += __shfl_xor(v, m, 32);
    return v;
}
__device__ __forceinline__ float warp_max(float v) {
#pragma unroll
    for (int m = 16; m >= 1; m >>= 1) v = fmaxf(v, __shfl_xor(v, m, 32));
    return v;
}

// ---------------------------------------------------------------------------
// C[M,Nc] = A[M,K] @ B[K,Nc], fp32, via V_WMMA_F32_16X16X4_F32.
// One wave per 16x16 output tile. M, K, Nc all multiples of 16 (and K of 4).
// A-frag (16x4 f32): lanes 0-15 hold M=lane, K={kk,kk+1}; lanes 16-31 hold
// M=lane-16, K={kk+2,kk+3}. B-frag mirrored (N across lanes, K split by half).
// C/D: VGPR r -> M = r + 8*half, N = lane%16.
// ---------------------------------------------------------------------------
__global__ void __launch_bounds__(256)
gat_wmma_gemm(const float* __restrict__ A, const float* __restrict__ B,
              float* __restrict__ C, int Mtiles, int K, int Nc) {
    const int wave   = (blockIdx.x * blockDim.x + threadIdx.x) >> 5;
    const int ntiles = Nc >> 4;
    if (wave >= Mtiles * ntiles) return;     // uniform per-wave exit
    const int tm   = wave / ntiles;
    const int tn   = wave % ntiles;
    const int lane = threadIdx.x & 31;
    const int half = lane >> 4;
    const int l16  = lane & 15;

    const float* Arow = A + (size_t)(tm * 16 + l16) * K;
    const float* Bcol = B + tn * 16 + l16;

    v8f acc = {};
    for (int kk = 0; kk < K; kk += 4) {
        const int k0 = kk + (half << 1);
        v2f a, b;
        a.x = Arow[k0];
        a.y = Arow[k0 + 1];
        b.x = Bcol[(size_t)k0 * Nc];
        b.y = Bcol[(size_t)(k0 + 1) * Nc];
        acc = __builtin_amdgcn_wmma_f32_16x16x4_f32(
            /*neg_a=*/false, a, /*neg_b=*/false, b,
            /*c_mod=*/(short)0, acc, /*reuse_a=*/false, /*reuse_b=*/false);
    }

    float* Crow = C + (size_t)(tm * 16 + (half << 3)) * Nc + tn * 16 + l16;
#pragma unroll
    for (int r = 0; r < 8; ++r) Crow[(size_t)r * Nc] = acc[r];
}

// alpha_s[n] = h[n,:]·a_s ; alpha_d[n] = h[n,:]·a_d   (one wave per node)
__global__ void node_alpha(const float* __restrict__ h,
                           const float* __restrict__ a_s,
                           const float* __restrict__ a_d,
                           float* __restrict__ out_s, float* __restrict__ out_d,
                           int N, int D) {
    const int node = (blockIdx.x * blockDim.x + threadIdx.x) >> 5;
    if (node >= N) return;
    const int lane = threadIdx.x & 31;
    const float* hr = h + (size_t)node * D;
    float ss = 0.f, sd = 0.f;
    for (int d = lane; d < D; d += 32) {
        float v = hr[d];
        ss += v * a_s[d];
        sd += v * a_d[d];
    }
    ss = warp_sum(ss);
    sd = warp_sum(sd);
    if (lane == 0) { out_s[node] = ss; out_d[node] = sd; }
}

__device__ __forceinline__ float edge_logit(int s, int d,
                                            const float* __restrict__ as,
                                            const float* __restrict__ ad) {
    float v = as[s] + ad[d];
    return v > 0.f ? v : NEG_SLOPE * v;
}

// order-preserving float<->uint encoding for atomicMax on floats
__device__ __forceinline__ unsigned enc_f(float f) {
    unsigned b = __float_as_uint(f);
    return (b & 0x80000000u) ? ~b : (b | 0x80000000u);
}
__device__ __forceinline__ float dec_f(unsigned e) {
    return __uint_as_float((e & 0x80000000u) ? (e ^ 0x80000000u) : ~e);
}

__global__ void edge_segmax(const int* __restrict__ src, const int* __restrict__ dst,
                            const float* __restrict__ as, const float* __restrict__ ad,
                            unsigned* __restrict__ menc, int E) {
    const int e = blockIdx.x * blockDim.x + threadIdx.x;
    if (e >= E) return;
    atomicMax(menc + dst[e], enc_f(edge_logit(src[e], dst[e], as, ad)));
}

__global__ void edge_expsum(const int* __restrict__ src, const int* __restrict__ dst,
                            const float* __restrict__ as, const float* __restrict__ ad,
                            const unsigned* __restrict__ menc,
                            float* __restrict__ denom, int E) {
    const int e = blockIdx.x * blockDim.x + threadIdx.x;
    if (e >= E) return;
    const int d = dst[e];
    float w = __expf(edge_logit(src[e], d, as, ad) - dec_f(menc[d]));
    unsafeAtomicAdd(denom + d, w);
}

// out[dst,:] += softmax-coef * h[src,:]  (one wave per edge, D/32 dims/lane)
template <int D>
__global__ void edge_aggregate(const int* __restrict__ src, const int* __restrict__ dst,
                               const float* __restrict__ as, const float* __restrict__ ad,
                               const unsigned* __restrict__ menc,
                               const float* __restrict__ denom,
                               const float* __restrict__ h,
                               float* __restrict__ out, int E) {
    const int e = (blockIdx.x * blockDim.x + threadIdx.x) >> 5;
    if (e >= E) return;
    const int lane = threadIdx.x & 31;
    const int s = src[e], d = dst[e];
    const float coef =
        __expf(edge_logit(s, d, as, ad) - dec_f(menc[d])) / denom[d];
    const float* hs = h + (size_t)s * D;
    float* od = out + (size_t)d * D;
#pragma unroll
    for (int k = lane; k < D; k += 32) unsafeAtomicAdd(od + k, coef * hs[k]);
}

__global__ void bias_relu_inplace(float* __restrict__ x, const float* __restrict__ b,
                                  int total, int Dmask) {
    const int i = blockIdx.x * blockDim.x + threadIdx.x;
    if (i >= total) return;
    x[i] = fmaxf(x[i] + b[i & Dmask], 0.f);
}

// log_softmax over 32 columns; one wave per node
__global__ void logsoftmax32(const float* __restrict__ agg, const float* __restrict__ b,
                             float* __restrict__ out, int N) {
    const int node = (blockIdx.x * blockDim.x + threadIdx.x) >> 5;
    if (node >= N) return;
    const int lane = threadIdx.x & 31;
    float v = agg[(size_t)node * 32 + lane] + b[lane];
    float m = warp_max(v);
    float s = warp_sum(__expf(v - m));
    out[(size_t)node * 32 + lane] = v - m - __logf(s);
}

extern "C" void kernel_launch(void* const* d_in, const int* in_sizes, int n_in,
                              void* d_out, int out_size, void* d_ws, size_t ws_size,
                              hipStream_t stream) {
    (void)n_in; (void)out_size; (void)ws_size;
    const float* x   = (const float*)d_in[0];
    const int*   ei  = (const int*)d_in[1];
    const float* W1  = (const float*)d_in[2];
    const float* as1 = (const float*)d_in[3];
    const float* ad1 = (const float*)d_in[4];
    const float* b1  = (const float*)d_in[5];
    const float* W2  = (const float*)d_in[6];
    const float* as2 = (const float*)d_in[7];
    const float* ad2 = (const float*)d_in[8];
    const float* b2  = (const float*)d_in[9];

    const int DH = 128, DO = 32;
    const int N = in_sizes[0] / DH;       // 100000
    const int E = in_sizes[1] / 2;        // 1,700,000 (incl. self-loops)
    const int* src = ei;
    const int* dst = ei + E;

    // ---- workspace layout (bytes) ----
    char* ws = (char*)d_ws;
    float*    h     = (float*)ws;                               // N*128
    float*    agg   = (float*)(ws + (size_t)N * DH * 4);        // N*128
    char*     tail  = ws + 2 * (size_t)N * DH * 4;
    float*    alS   = (float*)tail;                             // N
    float*    alD   = alS + N;                                  // N
    unsigned* menc  = (unsigned*)(alD + N);                     // N
    float*    denom = (float*)(menc + N);                       // N
    // layer-2 buffers recycle the h region (h1 dead after layer-1 aggregate)
    float* h2   = h;                 // N*32
    float* agg2 = h + (size_t)N * DO; // N*32

    const int nodeBlocks = (N + 7) / 8;        // 8 waves / 256-thread block
    const int edgeBlocks = (E + 255) / 256;
    const int edgeWaveBlocks = (E + 7) / 8;

    // ===== Layer 1 =====
    {
        int waves = (N / 16) * (DH / 16);
        gat_wmma_gemm<<<(waves + 7) / 8, 256, 0, stream>>>(x, W1, h, N / 16, DH, DH);
    }
    node_alpha<<<nodeBlocks, 256, 0, stream>>>(h, as1, ad1, alS, alD, N, DH);
    hipMemsetAsync(menc, 0, (size_t)N * 4, stream);
    hipMemsetAsync(denom, 0, (size_t)N * 4, stream);
    hipMemsetAsync(agg, 0, (size_t)N * DH * 4, stream);
    edge_segmax<<<edgeBlocks, 256, 0, stream>>>(src, dst, alS, alD, menc, E);
    edge_expsum<<<edgeBlocks, 256, 0, stream>>>(src, dst, alS, alD, menc, denom, E);
    edge_aggregate<128><<<edgeWaveBlocks, 256, 0, stream>>>(src, dst, alS, alD, menc,
                                                            denom, h, agg, E);
    bias_relu_inplace<<<((N * DH) + 255) / 256, 256, 0, stream>>>(agg, b1, N * DH, DH - 1);

    // ===== Layer 2 =====
    {
        int waves = (N / 16) * (DO / 16);
        gat_wmma_gemm<<<(waves + 7) / 8, 256, 0, stream>>>(agg, W2, h2, N / 16, DH, DO);
    }
    node_alpha<<<nodeBlocks, 256, 0, stream>>>(h2, as2, ad2, alS, alD, N, DO);
    hipMemsetAsync(menc, 0, (size_t)N * 4, stream);
    hipMemsetAsync(denom, 0, (size_t)N * 4, stream);
    hipMemsetAsync(agg2, 0, (size_t)N * DO * 4, stream);
    edge_segmax<<<edgeBlocks, 256, 0, stream>>>(src, dst, alS, alD, menc, E);
    edge_expsum<<<edgeBlocks, 256, 0, stream>>>(src, dst, alS, alD, menc, denom, E);
    edge_aggregate<32><<<edgeWaveBlocks, 256, 0, stream>>>(src, dst, alS, alD, menc,
                                                           denom, h2, agg2, E);
    logsoftmax32<<<nodeBlocks, 256, 0, stream>>>(agg2, b2, (float*)d_out, N);
}